// GraphSAGE_65326452572485
// MI455X (gfx1250) — compile-verified
//
#include <hip/hip_runtime.h>

typedef __attribute__((ext_vector_type(2))) float v2f;
typedef __attribute__((ext_vector_type(8))) float v8f;

#define NN   50000
#define EE   800000
#define D    128
#define DOUTN 64

// ---------------------------------------------------------------- zero fill
__global__ void zero_f32(float* __restrict__ p, int n) {
  int i = blockIdx.x * blockDim.x + threadIdx.x;
  int stride = gridDim.x * blockDim.x;
  for (; i < n; i += stride) p[i] = 0.0f;
}

// ---------------------------------------------------------------- degrees
__global__ void degree_kernel(const long long* __restrict__ ei,
                              float* __restrict__ cnt) {
  int e = blockIdx.x * blockDim.x + threadIdx.x;
  if (e < EE) {
    int dst = (int)ei[(size_t)EE + e];
    atomicAdd(&cnt[dst], 1.0f);
  }
}

// ------------------------------------------------- edge gather + scatter-add
// one wave (32 lanes) per edge, float4 per lane -> 128 floats per edge
__global__ void __launch_bounds__(256)
scatter_kernel(const long long* __restrict__ ei,
               const float* __restrict__ h,
               float* __restrict__ agg) {
  int tid = blockIdx.x * blockDim.x + threadIdx.x;
  int e = tid >> 5;
  if (e >= EE) return;
  int d0 = (tid & 31) << 2;
  int src = (int)ei[e];
  int dst = (int)ei[(size_t)EE + e];
  const float4 v = *(const float4*)(h + (size_t)src * D + d0);
  float* a = agg + (size_t)dst * D + d0;
  atomicAdd(a + 0, v.x);
  atomicAdd(a + 1, v.y);
  atomicAdd(a + 2, v.z);
  atomicAdd(a + 3, v.w);
}

// ------------------------------------------------- fused SAGE layer (WMMA)
// out = relu( (agg/max(cnt,1)) @ Wl + h @ Wr + bias )
// block = 256 threads = 8 waves; block owns 16 rows, wave w owns cols [16w,16w+16)
__global__ void __launch_bounds__(256)
sage_gemm_relu(const float* __restrict__ agg, const float* __restrict__ cnt,
               const float* __restrict__ h,
               const float* __restrict__ Wl, const float* __restrict__ Wr,
               const float* __restrict__ bias, float* __restrict__ out) {
  const int lane  = threadIdx.x & 31;
  const int wave  = threadIdx.x >> 5;          // 0..7 -> column tile
  const int r0    = blockIdx.x * 16;
  const int n16   = lane & 15;
  const int col   = wave * 16 + n16;           // B/C/D: N = lane%16
  const int row   = r0 + n16;                  // A: M = lane%16
  const int khalf = (lane >> 4) << 1;          // A: K = 2*(lane/16) + vgpr
  const float inv = 1.0f / fmaxf(cnt[row], 1.0f);

  const float* __restrict__ arow = agg + (size_t)row * D;
  const float* __restrict__ hrow = h   + (size_t)row * D;

  v8f acc = {};
#pragma unroll 4
  for (int kb = 0; kb < D; kb += 4) {
    const int k = kb + khalf;
    // A tiles: contiguous float2 along K (row-major features)
    v2f a0 = *(const v2f*)(arow + k);
    a0 *= inv;                                  // fold mean-normalization
    v2f a1 = *(const v2f*)(hrow + k);
    // B tiles: lane L holds B[K=2*(L/16)+v][N=L%16]
    v2f b0; b0.x = Wl[(size_t)k * D + col];     b0.y = Wl[(size_t)(k + 1) * D + col];
    v2f b1; b1.x = Wr[(size_t)k * D + col];     b1.y = Wr[(size_t)(k + 1) * D + col];
    acc = __builtin_amdgcn_wmma_f32_16x16x4_f32(false, a0, false, b0,
                                                (short)0, acc, false, false);
    acc = __builtin_amdgcn_wmma_f32_16x16x4_f32(false, a1, false, b1,
                                                (short)0, acc, false, false);
  }

  const float bc = bias[col];
#pragma unroll
  for (int i = 0; i < 8; ++i) {
    // C/D layout: VGPR i, lane L -> M = i + 8*(L/16), N = L%16
    const int orow = r0 + i + ((lane >> 4) << 3);
    out[(size_t)orow * D + col] = fmaxf(acc[i] + bc, 0.0f);
  }
}

// ------------------------------------------------- final linear + softmax
// block = 128 threads = 4 waves; 16 rows x 64 cols, softmax fused via LDS
__global__ void __launch_bounds__(128)
out_softmax(const float* __restrict__ h, const float* __restrict__ Wout,
            const float* __restrict__ bout, float* __restrict__ out) {
  __shared__ float tile[16][DOUTN + 4];

  const int lane  = threadIdx.x & 31;
  const int wave  = threadIdx.x >> 5;          // 0..3 -> column tile
  const int r0    = blockIdx.x * 16;
  const int n16   = lane & 15;
  const int col   = wave * 16 + n16;
  const int row   = r0 + n16;
  const int khalf = (lane >> 4) << 1;
  const float* __restrict__ hrow = h + (size_t)row * D;

  v8f acc = {};
#pragma unroll 4
  for (int kb = 0; kb < D; kb += 4) {
    const int k = kb + khalf;
    v2f a = *(const v2f*)(hrow + k);
    v2f b; b.x = Wout[(size_t)k * DOUTN + col];
           b.y = Wout[(size_t)(k + 1) * DOUTN + col];
    acc = __builtin_amdgcn_wmma_f32_16x16x4_f32(false, a, false, b,
                                                (short)0, acc, false, false);
  }

  const float bc = bout[col];
#pragma unroll
  for (int i = 0; i < 8; ++i)
    tile[i + ((lane >> 4) << 3)][col] = acc[i] + bc;

  __syncthreads();

  if (threadIdx.x < 16) {
    const int r = threadIdx.x;
    float m = -1e30f;
#pragma unroll
    for (int c = 0; c < DOUTN; ++c) m = fmaxf(m, tile[r][c]);
    float s = 0.0f;
#pragma unroll
    for (int c = 0; c < DOUTN; ++c) { float e = __expf(tile[r][c] - m); tile[r][c] = e; s += e; }
    const float invs = 1.0f / s;
    float* __restrict__ orow = out + (size_t)(r0 + r) * DOUTN;
#pragma unroll
    for (int c = 0; c < DOUTN; ++c) orow[c] = tile[r][c] * invs;
  }
}

// ---------------------------------------------------------------- launcher
extern "C" void kernel_launch(void* const* d_in, const int* in_sizes, int n_in,
                              void* d_out, int out_size, void* d_ws, size_t ws_size,
                              hipStream_t stream) {
  (void)in_sizes; (void)n_in; (void)out_size; (void)ws_size;

  const float*      x    = (const float*)d_in[0];
  const long long*  ei   = (const long long*)d_in[1];   // int64 edge_index [2,E]
  const float*      Wl0  = (const float*)d_in[2];
  const float*      Wr0  = (const float*)d_in[3];
  const float*      b0   = (const float*)d_in[4];
  const float*      Wl1  = (const float*)d_in[5];
  const float*      Wr1  = (const float*)d_in[6];
  const float*      b1   = (const float*)d_in[7];
  const float*      Wout = (const float*)d_in[8];
  const float*      bout = (const float*)d_in[9];
  float*            out  = (float*)d_out;

  float* ws  = (float*)d_ws;
  float* agg = ws;                          // N*D
  float* cnt = ws + (size_t)NN * D;         // N   (adjacent -> single zero pass)
  float* h   = cnt + NN;                    // N*D

  const int scatter_blocks = (EE * 32) / 256;   // 100000, exact
  const int gemm_blocks    = NN / 16;           // 3125, exact

  // layer 0
  zero_f32<<<2048, 256, 0, stream>>>(agg, NN * D + NN);
  degree_kernel<<<(EE + 255) / 256, 256, 0, stream>>>(ei, cnt);
  scatter_kernel<<<scatter_blocks, 256, 0, stream>>>(ei, x, agg);
  sage_gemm_relu<<<gemm_blocks, 256, 0, stream>>>(agg, cnt, x, Wl0, Wr0, b0, h);

  // layer 1 (agg reused; h updated in place: out[n] depends only on row n)
  zero_f32<<<2048, 256, 0, stream>>>(agg, NN * D);
  scatter_kernel<<<scatter_blocks, 256, 0, stream>>>(ei, h, agg);
  sage_gemm_relu<<<gemm_blocks, 256, 0, stream>>>(agg, cnt, h, Wl1, Wr1, b1, h);

  // output head + softmax
  out_softmax<<<gemm_blocks, 128, 0, stream>>>(h, Wout, bout, out);
}